// ConvModel_6536940224561
// MI455X (gfx1250) — compile-verified
//
#include <hip/hip_runtime.h>

typedef __attribute__((ext_vector_type(2))) float v2f;
typedef __attribute__((ext_vector_type(8))) float v8f;

// ---------------- atomics: force native fire-and-forget L2 f32 atomic ----------------
__device__ __forceinline__ void atomAddF32(float* p, float v) {
  asm volatile("global_atomic_add_f32 %0, %1, off" : : "v"(p), "v"(v) : "memory");
}

// ---------------- utility kernels ----------------
__global__ void zero4_kernel(float4* __restrict__ p, long long n4) {
  long long i = (long long)blockIdx.x * blockDim.x + threadIdx.x;
  if (i < n4) p[i] = make_float4(0.f, 0.f, 0.f, 0.f);
}

__global__ void degree_kernel(const int* __restrict__ dst, int* __restrict__ deg, int nE) {
  int i = blockIdx.x * blockDim.x + threadIdx.x;
  if (i < nE) atomicAdd(&deg[dst[i]], 1);
}

__global__ void invdeg_kernel(const int* __restrict__ deg, float* __restrict__ invd, int n) {
  int i = blockIdx.x * blockDim.x + threadIdx.x;
  if (i < n) invd[i] = 1.0f / fmaxf((float)deg[i], 1.0f);
}

// Pack W[K,N] row-major into Wp[K/2][Np][2]:  Wp[k2][c][i] = W[2*k2+i][c]
// (zero-padded to Np columns) so a lane's (b.x,b.y) WMMA B-pair is one b64 load.
__global__ void pack_w_kernel(const float* __restrict__ W, float* __restrict__ Wp,
                              int K, int N, int Np) {
  int i = blockIdx.x * blockDim.x + threadIdx.x;
  int total = (K >> 1) * Np;
  if (i >= total) return;
  int k2 = i / Np;
  int c = i - k2 * Np;
  float x0 = 0.f, x1 = 0.f;
  if (c < N) {
    x0 = W[(size_t)(2 * k2) * N + c];
    x1 = W[(size_t)(2 * k2 + 1) * N + c];
  }
  *(float2*)(Wp + (size_t)k2 * 2 * Np + 2 * c) = make_float2(x0, x1);
}

// scatter-sum of source features into destination rows: msum[dst] += h[src]
// one thread per (edge, float4 chunk); d is a power of two, logC = log2(d/4)
__global__ void scatter_sum_kernel(const float* __restrict__ h, const int* __restrict__ src,
                                   const int* __restrict__ dst, float* __restrict__ msum,
                                   int nE, int d, int logC) {
  long long idx = (long long)blockIdx.x * blockDim.x + threadIdx.x;
  long long total = (long long)nE << logC;
  if (idx >= total) return;
  int e = (int)(idx >> logC);
  int c = (int)(idx & ((1 << logC) - 1));
  int s = src[e];
  int t = dst[e];
  const float4 v = *(const float4*)(h + (size_t)s * d + 4 * c);
  float* p = msum + (size_t)t * d + 4 * c;
  atomAddF32(p + 0, v.x);
  atomAddF32(p + 1, v.y);
  atomAddF32(p + 2, v.z);
  atomAddF32(p + 3, v.w);
}

// ---------------- WMMA f32 GEMM ----------------
__device__ __forceinline__ v8f wmma4(v2f a, v2f b, v8f c) {
  return __builtin_amdgcn_wmma_f32_16x16x4_f32(false, a, false, b, (short)0, c, false, false);
}

// out[M,N] = [ h | (msum * inv) ] @ W(2d x N) + bias, optional ReLU.
// One wave computes a 32(M) x 64(N) tile: 8 accumulators, B fragments reused 2x.
// Wp is the packed weight [K/2][Np][2]; Np is N padded to a multiple of 64.
template <bool RELU>
__global__ __launch_bounds__(256) void sage_gemm_wmma(
    const float* __restrict__ h, const float* __restrict__ msum,
    const float* __restrict__ invd, const float* __restrict__ Wp,
    const float* __restrict__ bias, float* __restrict__ out, int M, int d, int N, int Np) {
  const int w = blockIdx.x * (blockDim.x >> 5) + ((int)threadIdx.x >> 5);
  const int tilesN = Np >> 6;
  const int tilesM = M >> 5;  // 32-row M tiles (M is a multiple of 32)
  const int tM = w / tilesN;
  const int tN = w - tM * tilesN;
  if (tM >= tilesM) return;  // wave-uniform exit; active waves keep EXEC all-ones

  const int m0 = tM << 5;
  const int n0 = tN << 6;
  const int lane = (int)threadIdx.x & 31;
  const int half = lane >> 4;  // 0: K pair {0,1}; 1: K pair {2,3}
  const int lcol = lane & 15;  // A row index / B-C column index
  const int r0 = m0 + lcol;
  const int r1 = r0 + 16;
  const float ir0 = invd[r0];
  const float ir1 = invd[r1];
  const float* h0 = h + (size_t)r0 * d;
  const float* h1 = h + (size_t)r1 * d;
  const float* s0 = msum + (size_t)r0 * d;
  const float* s1 = msum + (size_t)r1 * d;
  const size_t ldb = 2 * (size_t)Np;
  const int boff = 2 * (n0 + lcol);

  v8f acc0[4] = {}, acc1[4] = {};

  // K in [0, d): self features from h, packed W rows [0, d)
  for (int k0 = 0; k0 < d; k0 += 4) {
    const int ka = k0 + 2 * half;  // even
    v2f a0 = *(const v2f*)(h0 + ka);
    v2f a1 = *(const v2f*)(h1 + ka);
    const float* bp = Wp + (size_t)(ka >> 1) * ldb + boff;
#pragma unroll
    for (int j = 0; j < 4; ++j) {
      v2f b = *(const v2f*)(bp + 32 * j);
      acc0[j] = wmma4(a0, b, acc0[j]);
      acc1[j] = wmma4(a1, b, acc1[j]);
    }
  }
  // K in [d, 2d): neighbor mean fused as msum*inv, packed W rows [d, 2d)
  for (int k0 = 0; k0 < d; k0 += 4) {
    const int ka = k0 + 2 * half;
    v2f a0 = *(const v2f*)(s0 + ka);
    v2f a1 = *(const v2f*)(s1 + ka);
    a0.x *= ir0; a0.y *= ir0;
    a1.x *= ir1; a1.y *= ir1;
    const float* bp = Wp + (size_t)((d + ka) >> 1) * ldb + boff;
#pragma unroll
    for (int j = 0; j < 4; ++j) {
      v2f b = *(const v2f*)(bp + 32 * j);
      acc0[j] = wmma4(a0, b, acc0[j]);
      acc1[j] = wmma4(a1, b, acc1[j]);
    }
  }

  // Epilogue: C/D layout — lane half selects M base (+0/+8), VGPR r is M offset,
  // lcol is the N index in the 16-wide subtile. Guard only here (Wp is zero-padded).
#pragma unroll
  for (int j = 0; j < 4; ++j) {
    int cc = n0 + j * 16 + lcol;
    if (cc < N) {
      float bj = bias[cc];
#pragma unroll
      for (int r = 0; r < 8; ++r) {
        float v0 = acc0[j][r] + bj;
        float v1 = acc1[j][r] + bj;
        if (RELU) {
          v0 = fmaxf(v0, 0.0f);
          v1 = fmaxf(v1, 0.0f);
        }
        out[(size_t)(m0 + half * 8 + r) * N + cc] = v0;
        out[(size_t)(m0 + 16 + half * 8 + r) * N + cc] = v1;
      }
    }
  }
}

// ---------------- host launcher ----------------
extern "C" void kernel_launch(void* const* d_in, const int* in_sizes, int n_in, void* d_out,
                              int out_size, void* d_ws, size_t ws_size, hipStream_t stream) {
  (void)in_sizes; (void)n_in; (void)out_size; (void)ws_size;
  const float* x  = (const float*)d_in[0];
  const float* W1 = (const float*)d_in[1];
  const float* b1 = (const float*)d_in[2];
  const float* W2 = (const float*)d_in[3];
  const float* b2 = (const float*)d_in[4];
  const float* W3 = (const float*)d_in[5];
  const float* b3 = (const float*)d_in[6];
  const int* src  = (const int*)d_in[7];
  const int* dst  = (const int*)d_in[8];
  float* out = (float*)d_out;

  constexpr int M = 40000, E = 640000, INF = 128, HID = 256, OUTF = 47, OUTP = 64;

  float* ws = (float*)d_ws;
  int*   deg  = (int*)ws;                      // 40000 ints
  float* invd = ws + 40000;                    // 40000 floats
  float* msum = ws + 80000;                    // 40000*256 floats (reused per layer)
  float* h1   = msum + (size_t)M * HID;        // 40000*256
  float* h2   = h1 + (size_t)M * HID;          // 40000*256
  float* Wp1  = h2 + (size_t)M * HID;          // 256*256
  float* Wp2  = Wp1 + (size_t)(2 * INF) * HID; // 512*256
  float* Wp3  = Wp2 + (size_t)(2 * HID) * HID; // 512*64 (N padded 47 -> 64)

  const int T = 256;
  auto blocks = [](long long n, int t) { return (int)((n + t - 1) / t); };

  // pack weights (tiny; once per launch, deterministic)
  pack_w_kernel<<<blocks((2 * INF / 2) * (long long)HID, T), T, 0, stream>>>(W1, Wp1, 2 * INF, HID, HID);
  pack_w_kernel<<<blocks((2 * HID / 2) * (long long)HID, T), T, 0, stream>>>(W2, Wp2, 2 * HID, HID, HID);
  pack_w_kernel<<<blocks((2 * HID / 2) * (long long)OUTP, T), T, 0, stream>>>(W3, Wp3, 2 * HID, OUTF, OUTP);

  // degrees (graph identical for all 3 layers)
  zero4_kernel<<<blocks(M / 4, T), T, 0, stream>>>((float4*)deg, M / 4);
  degree_kernel<<<blocks(E, T), T, 0, stream>>>(dst, deg, E);
  invdeg_kernel<<<blocks(M, T), T, 0, stream>>>(deg, invd, M);

  // ---- layer 1: d = 128, N = 256, ReLU ----
  {
    long long n4 = (long long)M * INF / 4;
    zero4_kernel<<<blocks(n4, T), T, 0, stream>>>((float4*)msum, n4);
    long long tot = (long long)E << 5;  // d/4 = 32 chunks
    scatter_sum_kernel<<<blocks(tot, T), T, 0, stream>>>(x, src, dst, msum, E, INF, 5);
    int tiles = (M / 32) * (HID / 64);  // 5000 waves
    sage_gemm_wmma<true><<<blocks((long long)tiles * 32, T), T, 0, stream>>>(
        x, msum, invd, Wp1, b1, h1, M, INF, HID, HID);
  }
  // ---- layer 2: d = 256, N = 256, ReLU ----
  {
    long long n4 = (long long)M * HID / 4;
    zero4_kernel<<<blocks(n4, T), T, 0, stream>>>((float4*)msum, n4);
    long long tot = (long long)E << 6;  // 64 chunks
    scatter_sum_kernel<<<blocks(tot, T), T, 0, stream>>>(h1, src, dst, msum, E, HID, 6);
    int tiles = (M / 32) * (HID / 64);
    sage_gemm_wmma<true><<<blocks((long long)tiles * 32, T), T, 0, stream>>>(
        h1, msum, invd, Wp2, b2, h2, M, HID, HID, HID);
  }
  // ---- layer 3: d = 256, N = 47 (padded to 64), no ReLU ----
  {
    long long n4 = (long long)M * HID / 4;
    zero4_kernel<<<blocks(n4, T), T, 0, stream>>>((float4*)msum, n4);
    long long tot = (long long)E << 6;
    scatter_sum_kernel<<<blocks(tot, T), T, 0, stream>>>(h2, src, dst, msum, E, HID, 6);
    int tiles = (M / 32) * (OUTP / 64);  // 1250 waves
    sage_gemm_wmma<false><<<blocks((long long)tiles * 32, T), T, 0, stream>>>(
        h2, msum, invd, Wp3, b3, out, M, HID, OUTF, OUTP);
  }
}